// PureGCNLayer_46213848105055
// MI455X (gfx1250) — compile-verified
//
#include <hip/hip_runtime.h>
#include <hip/hip_bf16.h>
#include <stdint.h>

// Problem constants (match reference)
#define NODES 8192
#define DIN   256
#define DOUT  256
#define EDGES 262144

// LDS row stride in floats for the staged W tile. 260 gives bank index
// (4*nl + khalf + k) mod 64 across the wave -> all 64 word-banks distinct
// for ds_load_b64 (conflict-free), and keeps 8B/16B alignment.
#define LDS_STRIDE 260

typedef float v2f __attribute__((ext_vector_type(2)));
typedef float v8f __attribute__((ext_vector_type(8)));
typedef int   v4i __attribute__((vector_size(16)));   // matches builtin param

// ---------------------------------------------------------------------------
// Kernel 1: init workspace.
//   - zero 8MB dedup bitmask (2M words -> 512K uint4 stores)
//   - acc = x              (self-loop contribution, 2M floats -> 512K float4)
//   - deg = 1.0            (self-loop degree)
//   - cnt = 0
// ---------------------------------------------------------------------------
__global__ void k_init(uint4* __restrict__ mask4,
                       float* __restrict__ deg,
                       int*   __restrict__ cnt,
                       float4* __restrict__ acc4,
                       const float4* __restrict__ x4) {
    int tid = blockIdx.x * blockDim.x + threadIdx.x;   // 0..524287
    mask4[tid] = make_uint4(0u, 0u, 0u, 0u);
    acc4[tid]  = x4[tid];
    if (tid < NODES) deg[tid] = 1.0f;
    if (tid == 0)    *cnt = 0;
}

// ---------------------------------------------------------------------------
// Kernel 2: dedup edges via 8192x8192 bitmask. First claimer of bit (s,d)
// increments deg[s] and appends (s,d) to the compacted unique-edge list.
// ---------------------------------------------------------------------------
__global__ void k_dedup(const int* __restrict__ ei,
                        uint32_t* __restrict__ mask,
                        float* __restrict__ deg,
                        int*   __restrict__ cnt,
                        int2*  __restrict__ uniq) {
    int e = blockIdx.x * blockDim.x + threadIdx.x;
    if (e >= EDGES) return;
    int s = ei[e];             // edge_index[0, e]
    int d = ei[EDGES + e];     // edge_index[1, e]
    unsigned idx = (unsigned)s * (unsigned)NODES + (unsigned)d;  // < 2^26
    unsigned bit = 1u << (idx & 31u);
    unsigned old = atomicOr(&mask[idx >> 5], bit);
    if (!(old & bit)) {
        atomicAdd(&deg[s], 1.0f);
        int p = atomicAdd(cnt, 1);
        uniq[p] = make_int2(s, d);
    }
}

// ---------------------------------------------------------------------------
// Kernel 3: one wave32 per unique edge: acc[s][:] += x[d][:] (256 f32).
// x and acc both L2-resident; fully coalesced f32 atomics.
// ---------------------------------------------------------------------------
__global__ void k_scatter(const int2* __restrict__ uniq,
                          const int*  __restrict__ cnt,
                          const float* __restrict__ x,
                          float* __restrict__ acc) {
    int wave = (blockIdx.x * blockDim.x + threadIdx.x) >> 5;
    int lane = threadIdx.x & 31;
    if (wave >= *cnt) return;
    int2 sd = uniq[wave];
    const float* xs = x   + (size_t)sd.y * DIN;
    float*       as = acc + (size_t)sd.x * DIN;
#pragma unroll
    for (int j = 0; j < 8; ++j) {
        int c = lane + 32 * j;
        atomicAdd(&as[c], xs[c]);
    }
}

// ---------------------------------------------------------------------------
// Kernel 4: out[8192,256] = (acc/deg) @ W^T + b, fp32 WMMA (16x16x4).
// Block = 4 waves = 128(M) x 64(N). The 64KB W slice for the block's N-range
// is staged into LDS once (async-to-LDS b128 when available, ASYNCcnt-
// tracked), then each wave's 4 B fragments come from conflict-free
// ds_load_b64 with immediate offsets while 2 A fragments stream from global.
// 8 WMMAs per k-step; 1/deg row-normalization fused into A loads.
// ---------------------------------------------------------------------------
__global__ __launch_bounds__(128)
void k_gemm(const float* __restrict__ acc,
            const float* __restrict__ deg,
            const float* __restrict__ Wm,
            const float* __restrict__ bias,
            float* __restrict__ out) {
    __shared__ float wlds[64 * LDS_STRIDE];   // 66560 B

    const int tid  = threadIdx.x;
    const int lane = tid & 31;
    const int wv   = tid >> 5;                            // 0..3
    const int m_base = (blockIdx.x * 4 + wv) * 32;        // 256 M-tiles of 32
    const int n_base = blockIdx.y * 64;                   // 4 N-supertiles

    // ---- cooperative stage of W[n_base..n_base+64) x [0..256) into LDS ----
    {
        const float* gsrc = Wm + (size_t)n_base * DIN;
#if __has_builtin(__builtin_amdgcn_global_load_async_to_lds_b128)
#pragma unroll
        for (int i = 0; i < 32; ++i) {
            int linear = tid + i * 128;                   // 0..4095
            int n = linear >> 6;                          // row 0..63
            int c = (linear & 63) << 2;                   // float col, step 4
            const float* gp = gsrc + n * DIN + c;
            float*       lp = wlds + n * LDS_STRIDE + c;
            __builtin_amdgcn_global_load_async_to_lds_b128(
                (__attribute__((address_space(1))) v4i*)gp,
                (__attribute__((address_space(3))) v4i*)lp, 0, 0);
        }
#if __has_builtin(__builtin_amdgcn_s_wait_asynccnt)
        __builtin_amdgcn_s_wait_asynccnt(0);
#else
        asm volatile("s_wait_asynccnt 0x0" ::: "memory");
#endif
#else   // fallback: plain b128 load + ds_store_b128
#pragma unroll
        for (int i = 0; i < 32; ++i) {
            int linear = tid + i * 128;
            int n = linear >> 6;
            int c = (linear & 63) << 2;
            *(float4*)(wlds + n * LDS_STRIDE + c) =
                *(const float4*)(gsrc + n * DIN + c);
        }
#endif
    }
    __syncthreads();

    const int nl    = lane & 15;
    const int khalf = (lane >> 4) * 2;
    const int rowA  = m_base + nl;                        // A frag 0 rows
    const int rowB  = rowA + 16;                          // A frag 1 rows

    const float rdeg0 = 1.0f / deg[rowA];
    const float rdeg1 = 1.0f / deg[rowB];

    const float* a0p = acc + (size_t)rowA * DIN + khalf;
    const float* a1p = acc + (size_t)rowB * DIN + khalf;
    const float* lb  = wlds + nl * LDS_STRIDE + khalf;    // B frag base (LDS)

    v8f c00 = {}, c01 = {}, c02 = {}, c03 = {};           // rows m_base..+15
    v8f c10 = {}, c11 = {}, c12 = {}, c13 = {};           // rows m_base+16..+31

    for (int k = 0; k < DIN; k += 4) {
        v2f a0 = *(const v2f*)(a0p + k);
        v2f a1 = *(const v2f*)(a1p + k);
        a0 *= rdeg0;                                      // fused normalization
        a1 *= rdeg1;
        v2f b0 = *(const v2f*)(lb + k);                           // ds_load_b64
        v2f b1 = *(const v2f*)(lb + 16 * LDS_STRIDE + k);
        v2f b2 = *(const v2f*)(lb + 32 * LDS_STRIDE + k);
        v2f b3 = *(const v2f*)(lb + 48 * LDS_STRIDE + k);
        c00 = __builtin_amdgcn_wmma_f32_16x16x4_f32(false, a0, false, b0, (short)0, c00, false, false);
        c10 = __builtin_amdgcn_wmma_f32_16x16x4_f32(false, a1, false, b0, (short)0, c10, false, false);
        c01 = __builtin_amdgcn_wmma_f32_16x16x4_f32(false, a0, false, b1, (short)0, c01, false, false);
        c11 = __builtin_amdgcn_wmma_f32_16x16x4_f32(false, a1, false, b1, (short)0, c11, false, false);
        c02 = __builtin_amdgcn_wmma_f32_16x16x4_f32(false, a0, false, b2, (short)0, c02, false, false);
        c12 = __builtin_amdgcn_wmma_f32_16x16x4_f32(false, a1, false, b2, (short)0, c12, false, false);
        c03 = __builtin_amdgcn_wmma_f32_16x16x4_f32(false, a0, false, b3, (short)0, c03, false, false);
        c13 = __builtin_amdgcn_wmma_f32_16x16x4_f32(false, a1, false, b3, (short)0, c13, false, false);
    }

    // Epilogue: add bias, store per C/D layout.
    const int ocol   = lane & 15;
    const int rbase0 = m_base + ((lane >> 4) ? 8 : 0);
    const int rbase1 = rbase0 + 16;
    const float bv0 = bias[n_base + ocol];
    const float bv1 = bias[n_base + ocol + 16];
    const float bv2 = bias[n_base + ocol + 32];
    const float bv3 = bias[n_base + ocol + 48];
#pragma unroll
    for (int r = 0; r < 8; ++r) {
        float* o0 = out + (size_t)(rbase0 + r) * DOUT + n_base + ocol;
        o0[0]  = c00[r] + bv0;
        o0[16] = c01[r] + bv1;
        o0[32] = c02[r] + bv2;
        o0[48] = c03[r] + bv3;
        float* o1 = out + (size_t)(rbase1 + r) * DOUT + n_base + ocol;
        o1[0]  = c10[r] + bv0;
        o1[16] = c11[r] + bv1;
        o1[32] = c12[r] + bv2;
        o1[48] = c13[r] + bv3;
    }
}

// ---------------------------------------------------------------------------
// Host launcher. Workspace layout (bytes):
//   [0,        8388608)  dedup bitmask (8192*8192 bits)
//   [8388608,  8421376)  deg    float[8192]
//   [8421376,  8421632)  cnt    int (padded)
//   [8421632, 10518784)  uniq   int2[EDGES]
//   [10518784,18907392)  acc    float[8192*256]
// ---------------------------------------------------------------------------
extern "C" void kernel_launch(void* const* d_in, const int* in_sizes, int n_in,
                              void* d_out, int out_size, void* d_ws, size_t ws_size,
                              hipStream_t stream) {
    const float* x  = (const float*)d_in[0];
    const int*   ei = (const int*)d_in[1];
    const float* Wm = (const float*)d_in[2];
    const float* b  = (const float*)d_in[3];
    float* out = (float*)d_out;

    char* ws = (char*)d_ws;
    uint32_t* mask = (uint32_t*)(ws);
    float*    deg  = (float*)(ws + 8388608);
    int*      cnt  = (int*)(ws + 8421376);
    int2*     uniq = (int2*)(ws + 8421632);
    float*    acc  = (float*)(ws + 10518784);

    k_init<<<2048, 256, 0, stream>>>((uint4*)mask, deg, cnt, (float4*)acc,
                                     (const float4*)x);
    k_dedup<<<EDGES / 256, 256, 0, stream>>>(ei, mask, deg, cnt, uniq);
    k_scatter<<<EDGES / 8, 256, 0, stream>>>(uniq, cnt, x, acc);
    k_gemm<<<dim3(64, 4), 128, 0, stream>>>(acc, deg, Wm, b, out);
}